// GAT_51170240364591
// MI455X (gfx1250) — compile-verified
//
#include <hip/hip_runtime.h>

typedef __attribute__((ext_vector_type(2))) float v2f;
typedef __attribute__((ext_vector_type(8))) float v8f;

#define NEG_SLOPE 0.2f

// ---------------------------------------------------------------------------
// GEMM: H[M,COUT] = X[M,CIN] * W[CIN,COUT], fp32 via V_WMMA_F32_16X16X4_F32.
// W is staged in LDS *pre-swizzled into B fragments*: for k-step m (k=4m)
// and column c, LDS holds (W[4m][c],W[4m+1][c]) then (W[4m+2][c],W[4m+3][c])
// as adjacent float2s. A lane's fragment is then ONE aligned ds_load_b64:
//   addr = (m*COUT*2 + rsel*2 + (lane>>4)) v2f-elements
// Lanes 0-15 hit banks {4r,4r+1}, lanes 16-31 banks {4r+2,4r+3}: all 64 LDS
// banks, conflict-free, and tile t is a constant +256B immediate offset.
// A-frag (ISA 7.12.2, 32-bit A 16x4): lanes 0-15 hold K=0,1; lanes 16-31 K=2,3.
// ---------------------------------------------------------------------------
template <int CIN, int COUT>
__global__ __launch_bounds__(256) void gemm_wmma(const float* __restrict__ X,
                                                 const float* __restrict__ Wg,
                                                 float* __restrict__ H, int M) {
  __shared__ v2f Wswz[(CIN / 4) * COUT * 2];  // == CIN*COUT floats
  {
    constexpr int TOT = (CIN / 4) * COUT * 2;
    for (int i = threadIdx.x; i < TOT; i += 256) {
      int m = i / (COUT * 2);
      int r = i - m * (COUT * 2);
      int c = r >> 1;
      int hsel = r & 1;  // 0 -> rows 4m,4m+1 ; 1 -> rows 4m+2,4m+3
      const float* wp = Wg + (size_t)(4 * m + 2 * hsel) * COUT + c;
      v2f w;
      w.x = wp[0];
      w.y = wp[COUT];
      Wswz[i] = w;
    }
  }
  __syncthreads();

  const int lane = threadIdx.x & 31;
  const int wave = threadIdx.x >> 5;
  const int mtile = blockIdx.x * 8 + wave;
  if (mtile * 16 >= M) return;  // wave-uniform guard (EXEC stays all-ones)
  const int m0 = mtile * 16;

  constexpr int NT = COUT / 16;
  v8f acc[NT] = {};

  const int rsel = lane & 15;        // row (A) / col (B) within tile
  const int hiHalf = lane >> 4;      // 0: K=0,1 ; 1: K=2,3
  const float* xrow = X + (size_t)(m0 + rsel) * CIN + hiHalf * 2;
  const v2f* wbase = Wswz + rsel * 2 + hiHalf;

  for (int k = 0; k < CIN; k += 4) {
    v2f a = *reinterpret_cast<const v2f*>(xrow + k);  // A[r][k(+2) .. +1]
    const v2f* wb = wbase + (k >> 2) * (COUT * 2);
#pragma unroll
    for (int t = 0; t < NT; ++t) {
      v2f bf = wb[t * 32];  // one aligned ds_load_b64 per fragment
      acc[t] = __builtin_amdgcn_wmma_f32_16x16x4_f32(
          false, a, false, bf, (short)0, acc[t], false, false);
    }
  }

  // D layout: VGPR i, lanes 0-15 -> row m0+i; lanes 16-31 -> row m0+8+i.
  const int rbase = m0 + hiHalf * 8;
#pragma unroll
  for (int t = 0; t < NT; ++t) {
#pragma unroll
    for (int i = 0; i < 8; ++i) {
      H[(size_t)(rbase + i) * COUT + t * 16 + rsel] = acc[t][i];
    }
  }
}

// ---------------------------------------------------------------------------
// Per-node projections: als[n] = h[n].a_src, ald[n] = h[n].a_dst.
// One wave32 per node, shuffle reduction.
// ---------------------------------------------------------------------------
template <int C>
__global__ __launch_bounds__(256) void row_proj(const float* __restrict__ H,
                                                const float* __restrict__ asrc,
                                                const float* __restrict__ adst,
                                                float* __restrict__ als,
                                                float* __restrict__ ald, int Nn) {
  const int lane = threadIdx.x & 31;
  const int wave = threadIdx.x >> 5;
  const int n = blockIdx.x * 8 + wave;
  if (n >= Nn) return;
  const float* hrow = H + (size_t)n * C;
  float s = 0.f, d = 0.f;
#pragma unroll
  for (int c = lane; c < C; c += 32) {
    float hv = hrow[c];
    s += hv * asrc[c];
    d += hv * adst[c];
  }
#pragma unroll
  for (int off = 16; off > 0; off >>= 1) {
    s += __shfl_xor(s, off, 32);
    d += __shfl_xor(d, off, 32);
  }
  if (lane == 0) {
    als[n] = s;
    ald[n] = d;
  }
}

// Order-preserving float <-> u32 encoding so segment-max can use atomicMax
// on unsigned (zero-initialized buffer acts as -inf; every node has a
// self-loop so every segment is non-empty).
__device__ __forceinline__ unsigned fenc(float f) {
  unsigned u = __float_as_uint(f);
  return u ^ ((u >> 31) ? 0xFFFFFFFFu : 0x80000000u);
}
__device__ __forceinline__ float fdec(unsigned u) {
  unsigned v = u ^ ((u & 0x80000000u) ? 0x80000000u : 0xFFFFFFFFu);
  return __uint_as_float(v);
}

__device__ __forceinline__ void edge_sd(int i, int E, const int* __restrict__ src,
                                        const int* __restrict__ dst, int& s, int& d) {
  if (i < E) {
    s = src[i];
    d = dst[i];
  } else {  // implicit self-loop
    s = i - E;
    d = s;
  }
}

__global__ __launch_bounds__(256) void edge_max(const int* __restrict__ src,
                                                const int* __restrict__ dst,
                                                const float* __restrict__ als,
                                                const float* __restrict__ ald,
                                                unsigned* __restrict__ menc,
                                                int E, int Etot) {
  int i = blockIdx.x * 256 + threadIdx.x;
  if (i >= Etot) return;
  int s, d;
  edge_sd(i, E, src, dst, s, d);
  float e = als[s] + ald[d];
  e = (e > 0.f) ? e : NEG_SLOPE * e;
  atomicMax(&menc[d], fenc(e));
}

__global__ __launch_bounds__(256) void edge_expsum(const int* __restrict__ src,
                                                   const int* __restrict__ dst,
                                                   const float* __restrict__ als,
                                                   const float* __restrict__ ald,
                                                   const unsigned* __restrict__ menc,
                                                   float* __restrict__ ee,
                                                   float* __restrict__ denom,
                                                   int E, int Etot) {
  int i = blockIdx.x * 256 + threadIdx.x;
  if (i >= Etot) return;
  int s, d;
  edge_sd(i, E, src, dst, s, d);
  float e = als[s] + ald[d];
  e = (e > 0.f) ? e : NEG_SLOPE * e;
  float x = __expf(e - fdec(menc[d]));
  ee[i] = x;
  atomicAdd(&denom[d], x);
}

// out[dst] += alpha * h[src]; one wave32 per edge, C/32 floats per lane,
// fp32 atomics (L2-resident: h and out are ~51MB each, L2 is 192MB).
template <int C>
__global__ __launch_bounds__(256) void edge_aggregate(const int* __restrict__ src,
                                                      const int* __restrict__ dst,
                                                      const float* __restrict__ H,
                                                      const float* __restrict__ ee,
                                                      const float* __restrict__ denom,
                                                      float* __restrict__ out,
                                                      int E, int Etot) {
  const int lane = threadIdx.x & 31;
  const int wave = threadIdx.x >> 5;
  int i = blockIdx.x * 8 + wave;
  if (i >= Etot) return;
  int s, d;
  edge_sd(i, E, src, dst, s, d);
  float alpha = ee[i] / denom[d];
  constexpr int VPT = C / 32;
  const float* hs = H + (size_t)s * C + lane * VPT;
  float* od = out + (size_t)d * C + lane * VPT;
#pragma unroll
  for (int j = 0; j < VPT; ++j) atomicAdd(&od[j], alpha * hs[j]);
}

template <int C>
__global__ __launch_bounds__(256) void bias_act(float* __restrict__ io,
                                                const float* __restrict__ b,
                                                int total, int relu) {
  int i = blockIdx.x * 256 + threadIdx.x;
  if (i >= total) return;
  float v = io[i] + b[i & (C - 1)];
  if (relu) v = fmaxf(v, 0.f);
  io[i] = v;
}

// ---------------------------------------------------------------------------
extern "C" void kernel_launch(void* const* d_in, const int* in_sizes, int n_in,
                              void* d_out, int out_size, void* d_ws, size_t ws_size,
                              hipStream_t stream) {
  const float* x   = (const float*)d_in[0];
  const int*   ei  = (const int*)d_in[1];
  const float* W1  = (const float*)d_in[2];
  const float* as1 = (const float*)d_in[3];
  const float* ad1 = (const float*)d_in[4];
  const float* b1  = (const float*)d_in[5];
  const float* W2  = (const float*)d_in[6];
  const float* as2 = (const float*)d_in[7];
  const float* ad2 = (const float*)d_in[8];
  const float* b2  = (const float*)d_in[9];
  const float* W3  = (const float*)d_in[10];
  const float* as3 = (const float*)d_in[11];
  const float* ad3 = (const float*)d_in[12];
  const float* b3  = (const float*)d_in[13];

  const int N = in_sizes[0] / 128;
  const int E = in_sizes[1] / 2;
  const int Etot = E + N;
  const int* srcp = ei;
  const int* dstp = ei + E;

  float*    buf0  = (float*)d_ws;                 // h buffer  [N,128]
  float*    buf1  = buf0 + (size_t)N * 128;       // agg/input [N,128]
  float*    als   = buf1 + (size_t)N * 128;       // [N]
  float*    ald   = als + N;                      // [N]
  unsigned* menc  = (unsigned*)(ald + N);         // [N] encoded segment max
  float*    denom = (float*)(menc + N);           // [N]
  float*    ee    = denom + N;                    // [Etot]

  const int gGemm = (N / 16 + 7) / 8;
  const int gProj = (N + 7) / 8;
  const int gE    = (Etot + 255) / 256;
  const int gEagg = (Etot + 7) / 8;
  float* outF = (float*)d_out;

  // ---- Layer 1 : x -> buf0 (h) -> buf1 (relu(agg+b1)) ----
  gemm_wmma<128, 128><<<gGemm, 256, 0, stream>>>(x, W1, buf0, N);
  row_proj<128><<<gProj, 256, 0, stream>>>(buf0, as1, ad1, als, ald, N);
  hipMemsetAsync(menc, 0, (size_t)N * 4, stream);
  hipMemsetAsync(denom, 0, (size_t)N * 4, stream);
  hipMemsetAsync(buf1, 0, (size_t)N * 128 * 4, stream);
  edge_max<<<gE, 256, 0, stream>>>(srcp, dstp, als, ald, menc, E, Etot);
  edge_expsum<<<gE, 256, 0, stream>>>(srcp, dstp, als, ald, menc, ee, denom, E, Etot);
  edge_aggregate<128><<<gEagg, 256, 0, stream>>>(srcp, dstp, buf0, ee, denom, buf1, E, Etot);
  bias_act<128><<<(N * 128 + 255) / 256, 256, 0, stream>>>(buf1, b1, N * 128, 1);

  // ---- Layer 2 : buf1 -> buf0 (h) -> buf1 (relu(agg+b2)) ----
  gemm_wmma<128, 128><<<gGemm, 256, 0, stream>>>(buf1, W2, buf0, N);
  row_proj<128><<<gProj, 256, 0, stream>>>(buf0, as2, ad2, als, ald, N);
  hipMemsetAsync(menc, 0, (size_t)N * 4, stream);
  hipMemsetAsync(denom, 0, (size_t)N * 4, stream);
  hipMemsetAsync(buf1, 0, (size_t)N * 128 * 4, stream);
  edge_max<<<gE, 256, 0, stream>>>(srcp, dstp, als, ald, menc, E, Etot);
  edge_expsum<<<gE, 256, 0, stream>>>(srcp, dstp, als, ald, menc, ee, denom, E, Etot);
  edge_aggregate<128><<<gEagg, 256, 0, stream>>>(srcp, dstp, buf0, ee, denom, buf1, E, Etot);
  bias_act<128><<<(N * 128 + 255) / 256, 256, 0, stream>>>(buf1, b2, N * 128, 1);

  // ---- Layer 3 : buf1 -> buf0 (h, C=64) -> d_out (agg + b3) ----
  gemm_wmma<128, 64><<<gGemm, 256, 0, stream>>>(buf1, W3, buf0, N);
  row_proj<64><<<gProj, 256, 0, stream>>>(buf0, as3, ad3, als, ald, N);
  hipMemsetAsync(menc, 0, (size_t)N * 4, stream);
  hipMemsetAsync(denom, 0, (size_t)N * 4, stream);
  hipMemsetAsync(outF, 0, (size_t)N * 64 * 4, stream);
  edge_max<<<gE, 256, 0, stream>>>(srcp, dstp, als, ald, menc, E, Etot);
  edge_expsum<<<gE, 256, 0, stream>>>(srcp, dstp, als, ald, menc, ee, denom, E, Etot);
  edge_aggregate<64><<<gEagg, 256, 0, stream>>>(srcp, dstp, buf0, ee, denom, outF, E, Etot);
  bias_act<64><<<(N * 64 + 255) / 256, 256, 0, stream>>>(outF, b3, N * 64, 0);
}